// GNNAutoEncoder_83056077570823
// MI455X (gfx1250) — compile-verified
//
#include <hip/hip_runtime.h>

#define N_NODES   100000
#define ROW_TILES (N_NODES / 16)   // 6250, exact
#define WAVES_PER_BLOCK 8          // 256 threads = 8 wave32

typedef float v2f __attribute__((ext_vector_type(2)));
typedef float v8f __attribute__((ext_vector_type(8)));

// ---------------------------------------------------------------- utilities

__global__ void zero_kernel(float* __restrict__ p, size_t n) {
    size_t i = (size_t)blockIdx.x * blockDim.x + threadIdx.x;
    size_t stride = (size_t)gridDim.x * blockDim.x;
    for (; i < n; i += stride) p[i] = 0.0f;
}

__global__ void degree_kernel(const long long* __restrict__ ei, int E,
                              float* __restrict__ deg) {
    int e = blockIdx.x * blockDim.x + threadIdx.x;
    if (e >= E) return;
    int dst = (int)ei[(size_t)E + e];
    unsafeAtomicAdd(&deg[dst], 1.0f);   // global_atomic_add_f32, no return
}

__global__ void invdeg_kernel(float* __restrict__ deg, int n) {
    int i = blockIdx.x * blockDim.x + threadIdx.x;
    if (i >= n) return;
    deg[i] = 1.0f / fmaxf(deg[i], 1.0f);
}

// --------------------------------------------------- edge gather/scatter-add
// one thread per (edge, 4-feature chunk): float4 gather + 4 fp32 atomics.
// Mean-scale folded in here: agg accumulates x[src] * inv_deg[dst] directly,
// so the WMMA loop consumes A-fragments with no extra VALU work.

template <int F>
__global__ void scatter_kernel(const float* __restrict__ x,
                               const long long* __restrict__ ei, int E,
                               const float* __restrict__ invdeg,
                               float* __restrict__ agg) {
    constexpr int QF = F / 4;
    size_t gtid = (size_t)blockIdx.x * blockDim.x + threadIdx.x;
    size_t total = (size_t)E * QF;
    if (gtid >= total) return;
    int e  = (int)(gtid / QF);
    int f4 = (int)(gtid % QF) * 4;
    int src = (int)ei[e];
    int dst = (int)ei[(size_t)E + e];
    const float s = invdeg[dst];
    const float4 v = *(const float4*)(x + (size_t)src * F + f4);
    float* a = agg + (size_t)dst * F + f4;
    unsafeAtomicAdd(a + 0, v.x * s);
    unsafeAtomicAdd(a + 1, v.y * s);
    unsafeAtomicAdd(a + 2, v.z * s);
    unsafeAtomicAdd(a + 3, v.w * s);
}

// --------------------------------------------------------------- fused GEMM
// out = agg_mean @ Wl^T + x @ Wr^T + bl  (+ReLU), via V_WMMA_F32_16X16X4_F32.
// One wave computes a 16-row x (NCOL*16)-col strip: one A-fragment load feeds
// NCOL independent WMMA accumulation chains (4x less A traffic, no WMMA->WMMA
// RAW chain). Weights for the block's column strip are staged in LDS.

template <int FI, int FO, bool RELU>
__launch_bounds__(256)
__global__ void sage_gemm_kernel(const float* __restrict__ agg,   // pre-scaled mean
                                 const float* __restrict__ xin,
                                 const float* __restrict__ Wl,
                                 const float* __restrict__ Wr,
                                 const float* __restrict__ bl,
                                 float* __restrict__ out) {
    constexpr int NCOL = (FO / 16 < 4) ? (FO / 16) : 4;  // col tiles per wave
    constexpr int K2 = 2 * FI;                           // concatenated K: [Wl|Wr]
    __shared__ float smem[NCOL * 16 * K2];               // <= 64 KB
    const int tid = threadIdx.x;
    const int colBase = blockIdx.y * (NCOL * 16);

    for (int i = tid; i < NCOL * 16 * K2; i += 256) {
        int n = i / K2;
        int k = i - n * K2;
        smem[i] = (k < FI) ? Wl[(size_t)(colBase + n) * FI + k]
                           : Wr[(size_t)(colBase + n) * FI + (k - FI)];
    }
    __syncthreads();

    const int wave = tid >> 5;
    const int lane = tid & 31;
    const int rowTile = blockIdx.x * WAVES_PER_BLOCK + wave;
    if (rowTile >= ROW_TILES) return;      // uniform per wave: EXEC all-1s below

    const int r16 = lane & 15;             // A-row (M) and B-col (N) for this lane
    const int kk  = (lane >> 4) * 2;       // K sub-offset per WMMA fragment layout
    const size_t rowBase = (size_t)rowTile * 16;
    const size_t row = rowBase + r16;
    const float* __restrict__ arow = agg + row * FI;
    const float* __restrict__ xrow = xin + row * FI;

    v8f acc[NCOL];
    #pragma unroll
    for (int c = 0; c < NCOL; ++c) acc[c] = (v8f){};

    #pragma unroll 8
    for (int k0 = 0; k0 < FI; k0 += 4) {   // phase 1: agg_mean @ Wl^T
        v2f a;
        a[0] = arow[k0 + kk];
        a[1] = arow[k0 + kk + 1];
        #pragma unroll
        for (int c = 0; c < NCOL; ++c) {
            const float* bp = smem + (c * 16 + r16) * K2 + k0 + kk;
            v2f b; b[0] = bp[0]; b[1] = bp[1];
            acc[c] = __builtin_amdgcn_wmma_f32_16x16x4_f32(
                false, a, false, b, (short)0, acc[c], false, false);
        }
    }
    #pragma unroll 8
    for (int k0 = 0; k0 < FI; k0 += 4) {   // phase 2: x @ Wr^T
        v2f a;
        a[0] = xrow[k0 + kk];
        a[1] = xrow[k0 + kk + 1];
        #pragma unroll
        for (int c = 0; c < NCOL; ++c) {
            const float* bp = smem + (c * 16 + r16) * K2 + FI + k0 + kk;
            v2f b; b[0] = bp[0]; b[1] = bp[1];
            acc[c] = __builtin_amdgcn_wmma_f32_16x16x4_f32(
                false, a, false, b, (short)0, acc[c], false, false);
        }
    }

    #pragma unroll
    for (int c = 0; c < NCOL; ++c) {       // C layout: row=(lane<16?v:v+8), col=lane&15
        const int col = colBase + c * 16 + r16;
        const float bias = bl[col];
        #pragma unroll
        for (int v = 0; v < 8; ++v) {
            size_t orow = rowBase + ((lane < 16) ? v : v + 8);
            float val = acc[c][v] + bias;
            if (RELU) val = fmaxf(val, 0.0f);
            out[orow * FO + col] = val;
        }
    }
}

// ------------------------------------------------------------------- launch

extern "C" void kernel_launch(void* const* d_in, const int* in_sizes, int n_in,
                              void* d_out, int out_size, void* d_ws, size_t ws_size,
                              hipStream_t stream) {
    const float*     x   = (const float*)d_in[0];
    const long long* ei  = (const long long*)d_in[1];
    const float* Wl1 = (const float*)d_in[2];
    const float* bl1 = (const float*)d_in[3];
    const float* Wr1 = (const float*)d_in[4];
    const float* Wl2 = (const float*)d_in[5];
    const float* bl2 = (const float*)d_in[6];
    const float* Wr2 = (const float*)d_in[7];
    const float* Wl3 = (const float*)d_in[8];
    const float* bl3 = (const float*)d_in[9];
    const float* Wr3 = (const float*)d_in[10];
    const float* Wl4 = (const float*)d_in[11];
    const float* bl4 = (const float*)d_in[12];
    const float* Wr4 = (const float*)d_in[13];
    const int E = in_sizes[1] / 2;

    float* ws     = (float*)d_ws;
    float* invdeg = ws;                                  // N
    float* agg    = invdeg + N_NODES;                    // N*128 (max FI)
    float* h1     = agg + (size_t)N_NODES * 128;         // N*128
    float* z      = h1  + (size_t)N_NODES * 128;         // N*32
    float* h3     = z   + (size_t)N_NODES * 32;          // N*128
    float* xhat   = (float*)d_out;                       // N*64

    const int ZB = 2048;                                 // grid-stride zero blocks
    const int gemmRB = (ROW_TILES + WAVES_PER_BLOCK - 1) / WAVES_PER_BLOCK; // 782

    // ---- shared in-degree (constant across all 4 layers)
    zero_kernel<<<ZB, 256, 0, stream>>>(invdeg, (size_t)N_NODES);
    degree_kernel<<<(E + 255) / 256, 256, 0, stream>>>(ei, E, invdeg);
    invdeg_kernel<<<(N_NODES + 255) / 256, 256, 0, stream>>>(invdeg, N_NODES);

    // ---- layer 1: x[64] -> h1[128], ReLU          (NCOL=4 -> gridDim.y = 2)
    zero_kernel<<<ZB, 256, 0, stream>>>(agg, (size_t)N_NODES * 64);
    scatter_kernel<64><<<(int)(((size_t)E * 16 + 255) / 256), 256, 0, stream>>>(
        x, ei, E, invdeg, agg);
    sage_gemm_kernel<64, 128, true><<<dim3(gemmRB, 2), 256, 0, stream>>>(
        agg, x, Wl1, Wr1, bl1, h1);

    // ---- layer 2: h1[128] -> z[32]                (NCOL=2 -> gridDim.y = 1)
    zero_kernel<<<ZB, 256, 0, stream>>>(agg, (size_t)N_NODES * 128);
    scatter_kernel<128><<<(int)(((size_t)E * 32 + 255) / 256), 256, 0, stream>>>(
        h1, ei, E, invdeg, agg);
    sage_gemm_kernel<128, 32, false><<<dim3(gemmRB, 1), 256, 0, stream>>>(
        agg, h1, Wl2, Wr2, bl2, z);

    // ---- layer 3: z[32] -> h3[128], ReLU          (NCOL=4 -> gridDim.y = 2)
    zero_kernel<<<ZB, 256, 0, stream>>>(agg, (size_t)N_NODES * 32);
    scatter_kernel<32><<<(int)(((size_t)E * 8 + 255) / 256), 256, 0, stream>>>(
        z, ei, E, invdeg, agg);
    sage_gemm_kernel<32, 128, true><<<dim3(gemmRB, 2), 256, 0, stream>>>(
        agg, z, Wl3, Wr3, bl3, h3);

    // ---- layer 4: h3[128] -> x_hat[64]            (NCOL=4 -> gridDim.y = 1)
    zero_kernel<<<ZB, 256, 0, stream>>>(agg, (size_t)N_NODES * 128);
    scatter_kernel<128><<<(int)(((size_t)E * 32 + 255) / 256), 256, 0, stream>>>(
        h3, ei, E, invdeg, agg);
    sage_gemm_kernel<128, 64, false><<<dim3(gemmRB, 1), 256, 0, stream>>>(
        agg, h3, Wl4, Wr4, bl4, xhat);
}